// SAGE_37726992728910
// MI455X (gfx1250) — compile-verified
//
#include <hip/hip_runtime.h>
#include <math.h>

// f32 WMMA fragment types: A 16x4 (2 f32/lane), B 4x16 (2 f32/lane), C/D 16x16 (8 f32/lane).
typedef __attribute__((ext_vector_type(2))) float v2f;
typedef __attribute__((ext_vector_type(4))) float v4f;
typedef __attribute__((ext_vector_type(8))) float v8f;

#define N1c 123904
#define N2c 11264
#define N3c 1024
#define DIN 128
#define DH  256
#define DOUT 47

// ---------------- zero-fill (graph-capture-safe init) ----------------
__global__ __launch_bounds__(256) void zero_f32(float* __restrict__ p, long n) {
  long i = (long)blockIdx.x * blockDim.x + threadIdx.x;
  if (i < n) p[i] = 0.0f;
}

// ---------------- edge scatter: mean-aggregation numerator + counts ----------------
// One wave32 per edge. Non-temporal coalesced row gather (rows used ~once, keep L2
// for the agg buffer the atomics hammer), f32 atomic scatter-add.
__global__ __launch_bounds__(256) void scatter_mean(
    const float* __restrict__ x, const int* __restrict__ src,
    const int* __restrict__ dst, float* __restrict__ agg,
    float* __restrict__ cnt, int E, int D) {
  int wave = (int)(((long)blockIdx.x * blockDim.x + threadIdx.x) >> 5);
  int lane = threadIdx.x & 31;
  if (wave >= E) return;
  int s = src[wave];
  int d = dst[wave];
  const float* xr = x + (size_t)s * D;
  float* ar = agg + (size_t)d * D;
  for (int c = lane * 4; c < D; c += 128) {   // D is a multiple of 128
    v4f v = __builtin_nontemporal_load((const v4f*)(xr + c));
    atomicAdd(ar + c + 0, v.x);
    atomicAdd(ar + c + 1, v.y);
    atomicAdd(ar + c + 2, v.z);
    atomicAdd(ar + c + 3, v.w);
  }
  if (lane == 0) atomicAdd(cnt + d, 1.0f);
}

// ---------------- fused SAGE linear: out = (agg/cnt)@Wl + xt@Wr + b, optional ReLU ----------------
// One wave per 16x64 output strip: 4 C accumulators share one A-fragment pair per K-step,
// issuing 8 v_wmma_f32_16x16x4_f32 per iteration. K/N are compile-time so all W offsets
// become immediate-offset loads. OOB columns (N=47 layer) are handled by clamping the
// load address: D column j depends only on B column j, so garbage in clamped lanes never
// reaches a stored element.
template <int K, int N, bool RELU>
__global__ __launch_bounds__(256) void sage_gemm_wmma(
    const float* __restrict__ agg, const float* __restrict__ cnt,
    const float* __restrict__ xt,
    const float* __restrict__ Wl, const float* __restrict__ Wr,
    const float* __restrict__ bias, float* __restrict__ out, int M) {
  constexpr int tilesN  = (N + 15) / 16;
  constexpr int ngroups = (tilesN + 3) / 4;   // 16x64 strip = 4 n-tiles per wave
  const int tilesM = M >> 4;                  // M is a multiple of 16 for all layers
  int wave = (int)(((long)blockIdx.x * blockDim.x + threadIdx.x) >> 5);
  if (wave >= tilesM * ngroups) return;       // wave-uniform: EXEC all-1s for WMMA
  int tm = wave / ngroups;
  int tg = wave % ngroups;
  int lane = threadIdx.x & 31;
  int half = lane >> 4;
  int lm   = lane & 15;
  int arow = tm * 16 + lm;                    // A row for this lane (both halves)
  int koff = half * 2;                        // K sub-offset within the 4-wide step

  float invc = 1.0f / fmaxf(cnt[arow], 1.0f); // segment-mean divisor folded into A
  const float* aggp = agg + (size_t)arow * K + koff;
  const float* xp   = xt  + (size_t)arow * K + koff;

  int ncol[4];
  const float* wlp[4];
  const float* wrp[4];
  v8f acc[4];
#pragma unroll
  for (int g = 0; g < 4; ++g) {
    ncol[g] = tg * 64 + g * 16 + lm;
    int nc = ncol[g] < N ? ncol[g] : N - 1;   // clamp: always-valid load address
    wlp[g] = Wl + (size_t)koff * N + nc;
    wrp[g] = Wr + (size_t)koff * N + nc;
    acc[g] = (v8f){0.f, 0.f, 0.f, 0.f, 0.f, 0.f, 0.f, 0.f};
  }

#pragma unroll 2
  for (int kb = 0; kb < K; kb += 4) {
    float2 t1 = *(const float2*)(aggp + kb);
    float2 t2 = *(const float2*)(xp + kb);
    v2f a1, a2;
    a1.x = t1.x * invc; a1.y = t1.y * invc;
    a2.x = t2.x;        a2.y = t2.y;
#pragma unroll
    for (int g = 0; g < 4; ++g) {
      v2f bl, br;
      bl.x = wlp[g][(size_t)kb * N];
      bl.y = wlp[g][(size_t)(kb + 1) * N];
      br.x = wrp[g][(size_t)kb * N];
      br.y = wrp[g][(size_t)(kb + 1) * N];
      acc[g] = __builtin_amdgcn_wmma_f32_16x16x4_f32(false, a1, false, bl, (short)0, acc[g], false, false);
      acc[g] = __builtin_amdgcn_wmma_f32_16x16x4_f32(false, a2, false, br, (short)0, acc[g], false, false);
    }
  }

  // C/D layout: VGPR r -> rows r (lanes 0-15) / r+8 (lanes 16-31), lane%16 = column.
  int rbase = tm * 16 + half * 8;
#pragma unroll
  for (int g = 0; g < 4; ++g) {
    if (ncol[g] < N) {
      float bv = bias[ncol[g]];
#pragma unroll
      for (int r = 0; r < 8; ++r) {
        float v = acc[g][r] + bv;
        if (RELU) v = fmaxf(v, 0.f);
        out[(size_t)(rbase + r) * N + ncol[g]] = v;
      }
    }
  }
}

// ---------------- row-wise log_softmax (one wave per row, N=47) ----------------
__global__ __launch_bounds__(32) void log_softmax_k(
    const float* __restrict__ logits, float* __restrict__ out, int N) {
  int row = blockIdx.x;
  int lane = threadIdx.x;
  const float* r = logits + (size_t)row * N;
  float v0 = (lane < N)      ? r[lane]      : -INFINITY;
  float v1 = (lane + 32 < N) ? r[lane + 32] : -INFINITY;
  float m = fmaxf(v0, v1);
  for (int o = 16; o > 0; o >>= 1) m = fmaxf(m, __shfl_xor(m, o, 32));
  float e = ((lane < N) ? expf(v0 - m) : 0.f) + ((lane + 32 < N) ? expf(v1 - m) : 0.f);
  for (int o = 16; o > 0; o >>= 1) e += __shfl_xor(e, o, 32);
  float lse = logf(e) + m;
  if (lane < N)      out[(size_t)row * N + lane]      = v0 - lse;
  if (lane + 32 < N) out[(size_t)row * N + lane + 32] = v1 - lse;
}

extern "C" void kernel_launch(void* const* d_in, const int* in_sizes, int n_in,
                              void* d_out, int out_size, void* d_ws, size_t ws_size,
                              hipStream_t stream) {
  const float* x   = (const float*)d_in[0];
  const float* Wl0 = (const float*)d_in[1];
  const float* Wr0 = (const float*)d_in[2];
  const float* b0  = (const float*)d_in[3];
  const float* Wl1 = (const float*)d_in[4];
  const float* Wr1 = (const float*)d_in[5];
  const float* b1  = (const float*)d_in[6];
  const float* Wl2 = (const float*)d_in[7];
  const float* Wr2 = (const float*)d_in[8];
  const float* b2  = (const float*)d_in[9];
  const int* src0 = (const int*)d_in[10];
  const int* dst0 = (const int*)d_in[11];
  const int* src1 = (const int*)d_in[12];
  const int* dst1 = (const int*)d_in[13];
  const int* src2 = (const int*)d_in[14];
  const int* dst2 = (const int*)d_in[15];
  int E0 = in_sizes[10], E1 = in_sizes[12], E2 = in_sizes[14];

  // workspace layout (floats)
  float* ws = (float*)d_ws;
  long o = 0;
  float* agg0 = ws + o;  o += (long)N1c * DIN;
  float* cnt0 = ws + o;  o += N1c;
  float* h1   = ws + o;  o += (long)N1c * DH;
  float* agg1 = ws + o;  o += (long)N2c * DH;
  float* cnt1 = ws + o;  o += N2c;
  float* h2   = ws + o;  o += (long)N2c * DH;
  float* agg2 = ws + o;  o += (long)N3c * DH;
  float* cnt2 = ws + o;  o += N3c;
  float* logits = ws + o;

  // zero the accumulator + count regions every call (harness does not re-init)
  long z0 = (long)N1c * DIN + N1c;
  long z1 = (long)N2c * DH + N2c;
  long z2 = (long)N3c * DH + N3c;
  zero_f32<<<(int)((z0 + 255) / 256), 256, 0, stream>>>(agg0, z0);
  zero_f32<<<(int)((z1 + 255) / 256), 256, 0, stream>>>(agg1, z1);
  zero_f32<<<(int)((z2 + 255) / 256), 256, 0, stream>>>(agg2, z2);

  // ---- layer 0 ----
  scatter_mean<<<(int)(((long)E0 * 32 + 255) / 256), 256, 0, stream>>>(
      x, src0, dst0, agg0, cnt0, E0, DIN);
  {
    long waves = (long)(N1c / 16) * 4;  // ngroups = 16/4 = 4
    sage_gemm_wmma<DIN, DH, true><<<(int)((waves * 32 + 255) / 256), 256, 0, stream>>>(
        agg0, cnt0, x, Wl0, Wr0, b0, h1, N1c);
  }
  // ---- layer 1 ----
  scatter_mean<<<(int)(((long)E1 * 32 + 255) / 256), 256, 0, stream>>>(
      h1, src1, dst1, agg1, cnt1, E1, DH);
  {
    long waves = (long)(N2c / 16) * 4;
    sage_gemm_wmma<DH, DH, true><<<(int)((waves * 32 + 255) / 256), 256, 0, stream>>>(
        agg1, cnt1, h1, Wl1, Wr1, b1, h2, N2c);
  }
  // ---- layer 2 ----
  scatter_mean<<<(int)(((long)E2 * 32 + 255) / 256), 256, 0, stream>>>(
      h2, src2, dst2, agg2, cnt2, E2, DH);
  {
    long waves = (long)(N3c / 16) * 1;  // ngroups = ceil(3/4) = 1
    sage_gemm_wmma<DH, DOUT, false><<<(int)((waves * 32 + 255) / 256), 256, 0, stream>>>(
        agg2, cnt2, h2, Wl2, Wr2, b2, logits, N3c);
  }
  log_softmax_k<<<N3c, 32, 0, stream>>>(logits, (float*)d_out, DOUT);
}